// NADE_6777458393493
// MI455X (gfx1250) — compile-verified
//
#include <hip/hip_runtime.h>
#include <stdint.h>

// ---------------------------------------------------------------------------
// NADE log_prob scan for MI455X (gfx1250, wave32).
// B=4096 batch rows, D=784 dims, H=128 hidden.
// TWO batch rows per wave: lanes 0-15 -> row b, lanes 16-31 -> row b+1,
// each lane owns 8 hidden units of its row. The wave-uniform scalar tail
// (logit -> p -> logs) is SIMD-shared across the two rows (5.5 trans/row/step;
// trans pipe is the roofline). Weights staged in 32-row tiles into LDS by the
// Tensor Data Mover, double-buffered (s_wait_tensorcnt + barriers), scanned
// as two 16-step sub-chunks per tile (halves the sync rounds: 25 not 49).
// Reductions: fused DPP adds confined to 16-lane rows -> both halves reduce
// independently in the same instructions.
// ---------------------------------------------------------------------------

#define NADE_D 784
#define NADE_H 128
#define NADE_B 4096
#define TILE 32                            // weight rows per TDM tile
#define NTILES ((NADE_D + TILE - 1) / TILE)  // 25 (last tile: 16 real rows)
#define SUB 16                             // scan steps per sub-chunk
#define WPB 8                              // waves per block
#define BLOCK (WPB * 32)                   // 256 threads
#define ROWS_PER_BLOCK (WPB * 2)           // 16 batch rows per block

typedef unsigned int u32;
typedef u32 u32x4_t __attribute__((ext_vector_type(4)));
typedef int i32x4_t __attribute__((ext_vector_type(4)));
typedef int i32x8_t __attribute__((ext_vector_type(8)));

#if defined(__HIP_DEVICE_COMPILE__) &&                                     \
    __has_builtin(__builtin_amdgcn_tensor_load_to_lds) &&                  \
    __has_builtin(__builtin_amdgcn_s_wait_tensorcnt)
#define USE_TDM 1
#else
#define USE_TDM 0
#endif

#if defined(__HIP_DEVICE_COMPILE__) && __has_builtin(__builtin_amdgcn_tanhf)
#define USE_TANH 1
#define ASCALE 0.5f          // keep a half-scaled: h = 0.5*tanh(a/2)+0.5
#else
#define USE_TANH 0
#define ASCALE (-1.44269504088896340736f)  // keep a * -log2(e): h = rcp(1+2^a')
#endif

#define NEG_LOG2E (-1.44269504088896340736f)
#define LN2 (0.69314718055994530942f)
#define LN_EPS (-23.025850929940457f)      // ln(1e-10)

// DPP lane-shuffled add (no LDS traffic); ctrl must be a literal.
// All four stages used stay within 16-lane DPP rows -> per-half reduction.
#define DPP_ADD(v, ctrl)                                                    \
  ((v) + __int_as_float(__builtin_amdgcn_update_dpp(                        \
             0, __float_as_int(v), (ctrl), 0xf, 0xf, true)))

__device__ __forceinline__ float lane_bcast(float v, int srclane) {
  return __int_as_float(__builtin_amdgcn_readlane(__float_as_int(v), srclane));
}

// hidden-unit sigmoid from the pre-scaled accumulator
__device__ __forceinline__ float h_sigmoid(float apre) {
#if USE_TANH
  return fmaf(0.5f, __builtin_amdgcn_tanhf(apre), 0.5f);
#else
  return __builtin_amdgcn_rcpf(1.0f + __builtin_amdgcn_exp2f(apre));
#endif
}

#if USE_TDM
// One TDM 2D tile load: TILE rows x 128 f32 cols, row stride 128 floats,
// from global row0 into LDS at lds_byte_addr. rows_remaining bounds the
// tensor so rows past D are zero-filled (never read). D# per CDNA5 ISA ch.8
// (groups 0/1; trailing groups zero => 2D tensor). clang-23 lane: 6-arg
// builtin (u32x4 g0, i32x8 g1, i32x4, i32x4, i32x8, i32 cpol)
__device__ __forceinline__ void tdm_load_rows(const float* g_row0,
                                              u32 lds_byte_addr,
                                              int rows_remaining) {
  unsigned long long ga = (unsigned long long)(uintptr_t)g_row0;
  u32x4_t g0;
  g0[0] = 1u;                                             // count=1, user desc
  g0[1] = lds_byte_addr;                                  // lds_addr [63:32]
  g0[2] = (u32)ga;                                        // global_addr lo
  g0[3] = ((u32)(ga >> 32) & 0x01FFFFFFu) | (2u << 30);   // addr[56:32]|type=2
  i32x8_t g1;
  g1[0] = (int)(2u << 16);                 // data_size=2 (4 bytes), no extras
  g1[1] = (int)((u32)NADE_H << 16);        // tensor_dim0 = 128 (bits 79:48 lo)
  g1[2] = (int)(((u32)rows_remaining & 0xFFFFu) << 16);   // tensor_dim1 lo
  g1[3] = (int)((u32)NADE_H << 16);        // tensor_dim1 hi=0 | tile_dim0=128
  g1[4] = (int)TILE;                       // tile_dim1=TILE, tile_dim2=0
  g1[5] = (int)NADE_H;                     // tensor_dim0_stride = 128 elems
  g1[6] = 0;                               // stride hi | tensor_dim1_stride lo
  g1[7] = 0;
  i32x4_t gz4 = {0, 0, 0, 0};
  i32x8_t gz8 = {0, 0, 0, 0, 0, 0, 0, 0};
  __builtin_amdgcn_tensor_load_to_lds(g0, g1, gz4, gz4, gz8, 0);
}
#endif

__global__ __launch_bounds__(BLOCK) void nade_scan_kernel(
    const float* __restrict__ x, const float* __restrict__ b_enc,
    const float* __restrict__ b_dec, const float* __restrict__ w_enc,
    const float* __restrict__ w_dec, float* __restrict__ out) {
  __shared__ float s_we[2][TILE * NADE_H];   // 16 KB each buffer
  __shared__ float s_wd[2][TILE * NADE_H];   // total 64 KB LDS / block

  const int tid = threadIdx.x;
  const int lane = tid & 31;
  const int l15 = lane & 15;               // step slot / hidden-group index
  const int half = lane >> 4;              // which of the wave's 2 rows
  const int brow =
      (blockIdx.x * WPB + (tid >> 5)) * 2 + half;  // this lane's batch row

  // Pre-scaled accumulator: lane owns hidden 8*l15 .. 8*l15+7 of its row.
  float4 ahA = *(const float4*)(b_enc + (size_t)brow * NADE_H + l15 * 8);
  float4 ahB = *(const float4*)(b_enc + (size_t)brow * NADE_H + l15 * 8 + 4);
  ahA.x *= ASCALE; ahA.y *= ASCALE; ahA.z *= ASCALE; ahA.w *= ASCALE;
  ahB.x *= ASCALE; ahB.y *= ASCALE; ahB.z *= ASCALE; ahB.w *= ASCALE;
  float lp = 0.0f;   // distributed: lane accumulates steps with j == l15

#if USE_TDM
  if (tid < 32) {  // wave 0 drives the tensor DMA pipeline
    tdm_load_rows(w_enc, (u32)(uintptr_t)&s_we[0][0], NADE_D);
    tdm_load_rows(w_dec, (u32)(uintptr_t)&s_wd[0][0], NADE_D);
  }
#endif

  for (int tc = 0; tc < NTILES; ++tc) {
    const int t0 = tc * TILE;
#if USE_TDM
    const int buf = tc & 1;
    // Per-wave counter: free no-op for waves 1..7, real wait for wave 0.
    __builtin_amdgcn_s_wait_tensorcnt(0);                // tile tc landed
    __syncthreads();  // tile visible to all; all done reading buf^1 (tc-1)
    if (tid < 32 && (tc + 1) < NTILES) {                 // prefetch tile tc+1
      const int t1 = t0 + TILE;
      tdm_load_rows(w_enc + (size_t)t1 * NADE_H,
                    (u32)(uintptr_t)&s_we[buf ^ 1][0], NADE_D - t1);
      tdm_load_rows(w_dec + (size_t)t1 * NADE_H,
                    (u32)(uintptr_t)&s_wd[buf ^ 1][0], NADE_D - t1);
    }
#else
    const int buf = 0;
    __syncthreads();  // previous tile fully consumed
    for (int idx = tid; idx < TILE * NADE_H; idx += BLOCK) {
      size_t g = (size_t)t0 * NADE_H + idx;  // rows contiguous in memory
      s_we[0][idx] = (g < (size_t)NADE_D * NADE_H) ? w_enc[g] : 0.0f;
      s_wd[0][idx] = (g < (size_t)NADE_D * NADE_H) ? w_dec[g] : 0.0f;
    }
    __syncthreads();
#endif

    // Two 16-step scan sub-chunks per staged tile (one in the last tile).
    const int nsub = ((t0 + TILE) <= NADE_D) ? 2 : 1;
    for (int s = 0; s < nsub; ++s) {
      const int i0 = t0 + s * SUB;
      // Lane l15==j caches step i0+j's scalars FOR ITS ROW.
      float xv = x[(size_t)brow * NADE_D + i0 + l15];
      float bdv = b_dec[(size_t)brow * NADE_D + i0 + l15];
      if ((i0 + SUB) < NADE_D) {  // prefetch next sub-chunk's scalars
        __builtin_prefetch(x + (size_t)brow * NADE_D + i0 + SUB + l15, 0, 0);
        __builtin_prefetch(b_dec + (size_t)brow * NADE_D + i0 + SUB + l15,
                           0, 0);
      }
      float psel = 0.0f;
      const float* wdB = &s_wd[buf][(s * SUB) * NADE_H + l15 * 8];
      const float* weB = &s_we[buf][(s * SUB) * NADE_H + l15 * 8];

#pragma unroll
      for (int j = 0; j < SUB; ++j) {
        // Both halves read the same LDS addresses (broadcast-friendly).
        float4 wdA = *(const float4*)(wdB + j * NADE_H);
        float4 wdBv = *(const float4*)(wdB + j * NADE_H + 4);
        float4 weA = *(const float4*)(weB + j * NADE_H);
        float4 weBv = *(const float4*)(weB + j * NADE_H + 4);

        // h = sigmoid(a) for this lane's 8 hidden units; partial dot.
        float part = h_sigmoid(ahA.x) * wdA.x;
        part = fmaf(h_sigmoid(ahA.y), wdA.y, part);
        part = fmaf(h_sigmoid(ahA.z), wdA.z, part);
        part = fmaf(h_sigmoid(ahA.w), wdA.w, part);
        part = fmaf(h_sigmoid(ahB.x), wdBv.x, part);
        part = fmaf(h_sigmoid(ahB.y), wdBv.y, part);
        part = fmaf(h_sigmoid(ahB.z), wdBv.z, part);
        part = fmaf(h_sigmoid(ahB.w), wdBv.w, part);
        // Per-half 16-lane reduction (DPP rows are 16 lanes: halves stay
        // independent). Afterwards every lane holds ITS row's dot sum.
        part = DPP_ADD(part, 0xB1);   // quad_perm [1,0,3,2]   (xor 1)
        part = DPP_ADD(part, 0x4E);   // quad_perm [2,3,0,1]   (xor 2)
        part = DPP_ADD(part, 0x141);  // row_half_mirror       (xor 4 equiv)
        part = DPP_ADD(part, 0x140);  // row_mirror            (xor 8 equiv)

        // Own-lane candidate: every lane evaluates the p/log block with its
        // own bias/x; lanes with l15==j hold the true step-j result.
        float logit = part + bdv;
        float Lp = NEG_LOG2E * logit;
        float e = __builtin_amdgcn_exp2f(Lp);
        float t = 1.0f + e;
        float p = __builtin_amdgcn_rcpf(t);
        float u = __builtin_amdgcn_logf(t);
        float lnp = fmaxf(-LN2 * u, LN_EPS);         // safe_log(p)
        float ln1mp = fmaxf(LN2 * (Lp - u), LN_EPS); // safe_log(1-p)
        float lpi = fmaf(xv, lnp - ln1mp, ln1mp);    // own-x log-prob cand

        bool mine = (l15 == j);
        lp += mine ? lpi : 0.0f;     // lane j (per half) keeps step j
        psel = mine ? p : psel;

        // rank-1 update: both halves need their own x_j broadcast.
        float xj = (half == 0) ? lane_bcast(xv, j) : lane_bcast(xv, j + 16);
        float xjs = ASCALE * xj;
        ahA.x = fmaf(xjs, weA.x, ahA.x);
        ahA.y = fmaf(xjs, weA.y, ahA.y);
        ahA.z = fmaf(xjs, weA.z, ahA.z);
        ahA.w = fmaf(xjs, weA.w, ahA.w);
        ahB.x = fmaf(xjs, weBv.x, ahB.x);
        ahB.y = fmaf(xjs, weBv.y, ahB.y);
        ahB.z = fmaf(xjs, weBv.z, ahB.z);
        ahB.w = fmaf(xjs, weBv.w, ahB.w);
      }
      // cond_p writeback: lane l15 holds step i0+l15's p for its row.
      out[NADE_B + (size_t)brow * NADE_D + i0 + l15] = psel;
    }
  }

  // log_p is distributed across the 16 lanes of each half: reduce per half
  // (same DPP stages), then lanes 0 and 16 write their rows.
  lp = DPP_ADD(lp, 0xB1);
  lp = DPP_ADD(lp, 0x4E);
  lp = DPP_ADD(lp, 0x141);
  lp = DPP_ADD(lp, 0x140);
  if (l15 == 0) out[brow] = -lp;
}

extern "C" void kernel_launch(void* const* d_in, const int* in_sizes, int n_in,
                              void* d_out, int out_size, void* d_ws,
                              size_t ws_size, hipStream_t stream) {
  (void)in_sizes; (void)n_in; (void)out_size; (void)d_ws; (void)ws_size;
  const float* x     = (const float*)d_in[0];
  const float* b_enc = (const float*)d_in[1];
  const float* b_dec = (const float*)d_in[2];
  const float* w_enc = (const float*)d_in[3];   // (784,1,128) flat == (784,128)
  const float* w_dec = (const float*)d_in[4];   // (784,128,1) flat == (784,128)
  float* out = (float*)d_out;                   // [4096 | 4096*784]
  dim3 grid(NADE_B / ROWS_PER_BLOCK);           // 256 blocks x 256 threads
  hipLaunchKernelGGL(nade_scan_kernel, grid, dim3(BLOCK), 0, stream,
                     x, b_enc, b_dec, w_enc, w_dec, out);
}